// Encoder_40046275067940
// MI455X (gfx1250) — compile-verified
//
#include <hip/hip_runtime.h>
#include <hip/hip_bf16.h>

typedef __attribute__((ext_vector_type(2))) float v2f;
typedef __attribute__((ext_vector_type(8))) float v8f;

// Hardware f32 scatter-add (global_atomic_add_f32, no return -> STOREcnt,
// implicitly drained by S_ENDPGM's wait-idle).
__device__ __forceinline__ void atadd_f32(float* p, float v) {
  asm volatile("global_atomic_add_f32 %0, %1, off" :: "v"(p), "v"(v) : "memory");
}

// ---------------- degree / norm precompute ----------------
__global__ void k_deg(const int* __restrict__ dst, float* __restrict__ deg, int E) {
  int e = blockIdx.x * blockDim.x + threadIdx.x;
  if (e < E) atadd_f32(&deg[dst[e]], 1.0f);
}

__global__ void k_dis(float* __restrict__ dd, int n) {
  int i = blockIdx.x * blockDim.x + threadIdx.x;
  if (i < n) dd[i] = rsqrtf(dd[i] + 1.0f);  // +1 = self loop; deg>=1 always
}

__global__ void k_norm(const int* __restrict__ src, const int* __restrict__ dst,
                       const float* __restrict__ dis, float* __restrict__ nrm,
                       int E, int n) {
  int e = blockIdx.x * blockDim.x + threadIdx.x;
  if (e < E) {
    nrm[e] = dis[src[e]] * dis[dst[e]];
  } else if (e < E + n) {
    float d = dis[e - E];
    nrm[e] = d * d;                          // self loop
  }
}

// ---------------- WMMA f32 GEMM: out[N x OUTC] = A[N x 128] * W[128 x OUTC] ----------------
// One block = 16-row strip; 8 waves, one 16x16 n-tile each. K=128 -> 32 chained
// v_wmma_f32_16x16x4_f32. A strip staged in LDS (stride 132: 16B-aligned float4
// stores, 64-bank conflict-free b64 reads).
template<int OUTC>
__global__ __launch_bounds__(256) void k_gemm(const float* __restrict__ A,
                                              const float* __restrict__ W,
                                              float* __restrict__ out) {
  constexpr int K = 128;
  __shared__ float As[16 * 132];
  const int tid = threadIdx.x;
  const int m0  = blockIdx.x * 16;
  for (int i = tid; i < 16 * (K / 4); i += 256) {     // 512 float4, 2 per thread
    int r = i >> 5, c4 = i & 31;
    float4 v = reinterpret_cast<const float4*>(A + (size_t)(m0 + r) * K)[c4];
    *reinterpret_cast<float4*>(&As[r * 132 + c4 * 4]) = v;
  }
  __syncthreads();
  const int wave = tid >> 5;
  const int lane = tid & 31;
  const int half = lane >> 4;
  const int mn   = lane & 15;
  const int n0   = wave * 16;
  if (n0 >= OUTC) return;                              // wave-uniform
  v8f c = {};
  const float* wp = W + n0 + mn;
  const float* ap = &As[mn * 132 + 2 * half];
#pragma unroll
  for (int k0 = 0; k0 < K; k0 += 4) {
    v2f a, b;
    a.x = ap[k0];                                      // A[m][k0+2*half]
    a.y = ap[k0 + 1];
    b.x = wp[(size_t)(k0 + 2 * half) * OUTC];          // W[k][n]
    b.y = wp[(size_t)(k0 + 2 * half + 1) * OUTC];
    c = __builtin_amdgcn_wmma_f32_16x16x4_f32(false, a, false, b, (short)0, c,
                                              false, false);
  }
#pragma unroll
  for (int v = 0; v < 8; ++v)
    out[(size_t)(m0 + v + 8 * half) * OUTC + n0 + mn] = c[v];
}

// Dual-head GEMM (OUTC=64 each): waves 0-3 -> Wa/oa, waves 4-7 -> Wb/ob; one
// shared LDS A load.
__global__ __launch_bounds__(256) void k_gemm_dual(const float* __restrict__ A,
                                                   const float* __restrict__ Wa,
                                                   const float* __restrict__ Wb,
                                                   float* __restrict__ oa,
                                                   float* __restrict__ ob) {
  constexpr int K = 128, OUTC = 64;
  __shared__ float As[16 * 132];
  const int tid = threadIdx.x;
  const int m0  = blockIdx.x * 16;
  for (int i = tid; i < 16 * (K / 4); i += 256) {
    int r = i >> 5, c4 = i & 31;
    float4 v = reinterpret_cast<const float4*>(A + (size_t)(m0 + r) * K)[c4];
    *reinterpret_cast<float4*>(&As[r * 132 + c4 * 4]) = v;
  }
  __syncthreads();
  const int wave = tid >> 5;
  const int lane = tid & 31;
  const int half = lane >> 4;
  const int mn   = lane & 15;
  const float* W  = (wave < 4) ? Wa : Wb;
  float*      out = (wave < 4) ? oa : ob;
  const int n0 = (wave & 3) * 16;
  v8f c = {};
  const float* wp = W + n0 + mn;
  const float* ap = &As[mn * 132 + 2 * half];
#pragma unroll
  for (int k0 = 0; k0 < K; k0 += 4) {
    v2f a, b;
    a.x = ap[k0];
    a.y = ap[k0 + 1];
    b.x = wp[(size_t)(k0 + 2 * half) * OUTC];
    b.y = wp[(size_t)(k0 + 2 * half + 1) * OUTC];
    c = __builtin_amdgcn_wmma_f32_16x16x4_f32(false, a, false, b, (short)0, c,
                                              false, false);
  }
#pragma unroll
  for (int v = 0; v < 8; ++v)
    out[(size_t)(m0 + v + 8 * half) * OUTC + n0 + mn] = c[v];
}

// ---------------- propagation: agg[dst] += norm * t[src] ----------------
// One wave32 per (edge or self-loop). C=128: float4/lane; C=64: float2/lane.
template<int C>
__global__ void k_prop(const float* __restrict__ t, float* __restrict__ agg,
                       const int* __restrict__ src, const int* __restrict__ dst,
                       const float* __restrict__ nrm, int E, int Etot) {
  const int gw   = (int)((blockIdx.x * blockDim.x + threadIdx.x) >> 5);
  const int lane = threadIdx.x & 31;
  if (gw >= Etot) return;
  int s, d;
  if (gw < E) { s = src[gw]; d = dst[gw]; } else { s = d = gw - E; }
  const float w = nrm[gw];
  if constexpr (C == 128) {
    const float4 v = reinterpret_cast<const float4*>(t + (size_t)s * C)[lane];
    float* o = agg + (size_t)d * C + lane * 4;
    atadd_f32(o + 0, w * v.x);
    atadd_f32(o + 1, w * v.y);
    atadd_f32(o + 2, w * v.z);
    atadd_f32(o + 3, w * v.w);
  } else {
    const float2 v = reinterpret_cast<const float2*>(t + (size_t)s * C)[lane];
    float* o = agg + (size_t)d * C + lane * 2;
    atadd_f32(o + 0, w * v.x);
    atadd_f32(o + 1, w * v.y);
  }
}

// ---------------- post-aggregation bias + activation ----------------
// ACT: 0=selu, 1=silu, 2=log_sigmoid, 3=bias only
template<int C, int ACT>
__global__ void k_bias_act(float* __restrict__ h, const float* __restrict__ b,
                           int total) {
  int i = blockIdx.x * blockDim.x + threadIdx.x;
  if (i >= total) return;
  float x = h[i] + b[i & (C - 1)];
  float y;
  if constexpr (ACT == 0) {
    const float alpha = 1.6732632423543772f, scale = 1.0507009873554805f;
    y = x > 0.f ? scale * x : scale * alpha * (expf(x) - 1.f);
  } else if constexpr (ACT == 1) {
    y = x / (1.f + expf(-x));
  } else if constexpr (ACT == 2) {
    y = fminf(x, 0.f) - log1pf(expf(-fabsf(x)));   // stable log(sigmoid(x))
  } else {
    y = x;
  }
  h[i] = y;
}

extern "C" void kernel_launch(void* const* d_in, const int* in_sizes, int n_in,
                              void* d_out, int out_size, void* d_ws, size_t ws_size,
                              hipStream_t stream) {
  const float* x   = (const float*)d_in[0];
  const int*   ei  = (const int*)d_in[1];
  const float* W0  = (const float*)d_in[2];
  const float* b0  = (const float*)d_in[3];
  const float* W1  = (const float*)d_in[4];
  const float* b1  = (const float*)d_in[5];
  const float* W2  = (const float*)d_in[6];
  const float* b2  = (const float*)d_in[7];
  const float* W3  = (const float*)d_in[8];
  const float* b3  = (const float*)d_in[9];
  const float* Wmu = (const float*)d_in[10];
  const float* bmu = (const float*)d_in[11];
  const float* Wlv = (const float*)d_in[12];
  const float* blv = (const float*)d_in[13];

  const int C  = 128, CO = 64;
  const int N    = in_sizes[0] / C;
  const int E    = in_sizes[1] / 2;
  const int Etot = E + N;
  const int* srcp = ei;        // edge_index[0]
  const int* dstp = ei + E;    // edge_index[1]

  float* ws   = (float*)d_ws;
  float* dis  = ws;                        // N floats (deg -> dis in place)
  float* nrm  = dis + N;                   // Etot floats
  float* bufA = nrm + Etot;                // N*128
  float* bufB = bufA + (size_t)N * C;      // N*128

  float* out_mu = (float*)d_out;
  float* out_lv = out_mu + (size_t)N * CO;

  const int B = 256;
  const int gGemm  = N / 16;                          // N % 16 == 0
  const int gProp  = (Etot * 32 + B - 1) / B;
  const int gAct   = (N * C + B - 1) / B;
  const int gAct64 = (N * CO + B - 1) / B;

  // ---- gcn_norm ----
  hipMemsetAsync(dis, 0, (size_t)N * sizeof(float), stream);
  k_deg<<<(E + B - 1) / B, B, 0, stream>>>(dstp, dis, E);
  k_dis<<<(N + B - 1) / B, B, 0, stream>>>(dis, N);
  k_norm<<<(Etot + B - 1) / B, B, 0, stream>>>(srcp, dstp, dis, nrm, E, N);

  // ---- layer 0: selu ----
  k_gemm<128><<<gGemm, B, 0, stream>>>(x, W0, bufA);
  hipMemsetAsync(bufB, 0, (size_t)N * C * sizeof(float), stream);
  k_prop<128><<<gProp, B, 0, stream>>>(bufA, bufB, srcp, dstp, nrm, E, Etot);
  k_bias_act<128, 0><<<gAct, B, 0, stream>>>(bufB, b0, N * C);

  // ---- layer 1: silu ----
  k_gemm<128><<<gGemm, B, 0, stream>>>(bufB, W1, bufA);
  hipMemsetAsync(bufB, 0, (size_t)N * C * sizeof(float), stream);
  k_prop<128><<<gProp, B, 0, stream>>>(bufA, bufB, srcp, dstp, nrm, E, Etot);
  k_bias_act<128, 1><<<gAct, B, 0, stream>>>(bufB, b1, N * C);

  // ---- layer 2: silu ----
  k_gemm<128><<<gGemm, B, 0, stream>>>(bufB, W2, bufA);
  hipMemsetAsync(bufB, 0, (size_t)N * C * sizeof(float), stream);
  k_prop<128><<<gProp, B, 0, stream>>>(bufA, bufB, srcp, dstp, nrm, E, Etot);
  k_bias_act<128, 1><<<gAct, B, 0, stream>>>(bufB, b2, N * C);

  // ---- layer 3: log_sigmoid ----
  k_gemm<128><<<gGemm, B, 0, stream>>>(bufB, W3, bufA);
  hipMemsetAsync(bufB, 0, (size_t)N * C * sizeof(float), stream);
  k_prop<128><<<gProp, B, 0, stream>>>(bufA, bufB, srcp, dstp, nrm, E, Etot);
  k_bias_act<128, 2><<<gAct, B, 0, stream>>>(bufB, b3, N * C);

  // ---- heads: mu / logvar ----
  float* tmu = bufA;
  float* tlv = bufA + (size_t)N * CO;
  k_gemm_dual<<<gGemm, B, 0, stream>>>(bufB, Wmu, Wlv, tmu, tlv);
  hipMemsetAsync(d_out, 0, (size_t)2 * N * CO * sizeof(float), stream);
  k_prop<64><<<gProp, B, 0, stream>>>(tmu, out_mu, srcp, dstp, nrm, E, Etot);
  k_prop<64><<<gProp, B, 0, stream>>>(tlv, out_lv, srcp, dstp, nrm, E, Etot);
  k_bias_act<64, 3><<<gAct64, B, 0, stream>>>(out_mu, bmu, N * CO);
  k_bias_act<64, 3><<<gAct64, B, 0, stream>>>(out_lv, blv, N * CO);
}